// ClothLinearFusion_74045236183240
// MI455X (gfx1250) — compile-verified
//
#include <hip/hip_runtime.h>
#include <hip/hip_bf16.h>

// Problem constants (from the reference)
#define BATCH    128
#define KDIM     64
#define CLOTH_CH 1024
#define BODY_CH  1024

typedef __attribute__((ext_vector_type(2))) float v2f;
typedef __attribute__((ext_vector_type(4))) float v4f;
typedef __attribute__((ext_vector_type(8))) float v8f;

// ---------------------------------------------------------------------------
// Kernel 1: c_sum[b,i] = sum_k cloth[b,k,i]
// Pure streaming reduction of 33.5 MB -> 0.5 MB (the HBM-bound part of the
// op; floor ~1.5us at 23.3 TB/s). 128-bit non-temporal loads: cloth is
// single-use, keep f/body/csum resident in L2 instead.
// One thread per float4 of output: 128*1024/4 = 32768 threads.
// ---------------------------------------------------------------------------
__global__ void __launch_bounds__(256)
cloth_csum_kernel(const float* __restrict__ cloth, float* __restrict__ csum)
{
    const int n4  = (BATCH * CLOTH_CH) / 4;
    int idx = blockIdx.x * blockDim.x + threadIdx.x;
    if (idx >= n4) return;

    const int c4 = CLOTH_CH / 4;        // 256 float4 per channel row
    int b  = idx / c4;
    int i4 = idx % c4;

    const v4f* base = (const v4f*)cloth + (size_t)b * (KDIM * c4) + i4;

    v4f s = {0.f, 0.f, 0.f, 0.f};
#pragma unroll 8
    for (int k = 0; k < KDIM; ++k) {
        v4f v = __builtin_nontemporal_load(base + k * c4);
        s += v;
    }
    ((v4f*)csum)[idx] = s;
}

// ---------------------------------------------------------------------------
// Kernel 2: out[b,i] = csum[b,i] * sum_j body[b,j] * f[i,j]
// GEMM M=128 (batch), N=1024 (cloth ch), K=1024 (body ch) via
// V_WMMA_F32_16X16X4_F32 (exact f32 path). A[m,k] = body[m,k]; B[k,n] = f[n,k].
//
// ISA VGPR layouts (CDNA5 ISA §7.12.2, wave32):
//   A 16x4 f32 : lanes 0-15 = M, {v0,v1} = K{0,1}; lanes 16-31 = M, K{2,3}
//   B 4x16 f32 : lanes 0-15 = N, {v0,v1} = K{0,1}; lanes 16-31 = N, K{2,3}
//   -> per lane, per k-step of 4: one aligned float2 per 16-row tile of
//      body / f. Both contiguous in K.
//   D 16x16 f32: VGPR r = row (r + 8*half), lanes(0-15) = col.
//
// Register blocking: each wave computes a 32x32 output tile (2 M-tiles x
// 2 N-tiles, 4 accumulators). Each A/B load feeds two WMMAs -> 1 b64 load
// per WMMA (halves L2 traffic to ~32 MB) and 4 independent C chains for
// WMMA pipelining. 4x32 = 128 waves = 16 blocks x 256 threads.
// ---------------------------------------------------------------------------
__global__ void __launch_bounds__(256)
fused_gemm_wmma_kernel(const float* __restrict__ body,  // (128,1024)
                       const float* __restrict__ f,     // (1024,1024) [i][j]
                       const float* __restrict__ csum,  // (128,1024)
                       float* __restrict__ out)         // (128,1024)
{
    const int wave    = (blockIdx.x * blockDim.x + threadIdx.x) >> 5;
    const int lane    = threadIdx.x & 31;
    const int lane_lo = lane & 15;
    const int half    = lane >> 4;
    const int khalf   = half << 1;            // 0 or 2

    const int mt = wave >> 5;                 // 0..3   (32-row batch tile)
    const int nt = wave & 31;                 // 0..31  (32-col channel tile)
    const int m0 = mt * 32;
    const int n0 = nt * 32;

    // Per-lane base pointers; offsets stay 8-byte aligned (khalf even, k%4==0).
    const float* aptr0 = body + (size_t)(m0 +  0 + lane_lo) * BODY_CH + khalf;
    const float* aptr1 = body + (size_t)(m0 + 16 + lane_lo) * BODY_CH + khalf;
    const float* bptr0 = f    + (size_t)(n0 +  0 + lane_lo) * BODY_CH + khalf;
    const float* bptr1 = f    + (size_t)(n0 + 16 + lane_lo) * BODY_CH + khalf;

    v8f c00 = {0.f, 0.f, 0.f, 0.f, 0.f, 0.f, 0.f, 0.f};
    v8f c01 = c00, c10 = c00, c11 = c00;

#pragma unroll 8
    for (int k = 0; k < BODY_CH; k += 4) {
        v2f a0 = *(const v2f*)(aptr0 + k);
        v2f a1 = *(const v2f*)(aptr1 + k);
        v2f b0 = *(const v2f*)(bptr0 + k);
        v2f b1 = *(const v2f*)(bptr1 + k);
        // 8 args: (neg_a, A, neg_b, B, c_mod, C, reuse_a, reuse_b)
        c00 = __builtin_amdgcn_wmma_f32_16x16x4_f32(false, a0, false, b0, (short)0, c00, false, false);
        c01 = __builtin_amdgcn_wmma_f32_16x16x4_f32(false, a0, false, b1, (short)0, c01, false, false);
        c10 = __builtin_amdgcn_wmma_f32_16x16x4_f32(false, a1, false, b0, (short)0, c10, false, false);
        c11 = __builtin_amdgcn_wmma_f32_16x16x4_f32(false, a1, false, b1, (short)0, c11, false, false);
    }

    // Epilogue: fuse the elementwise csum multiply. Row per VGPR, cols across
    // lanes -> 64B coalesced accesses per half-wave.
#pragma unroll
    for (int r = 0; r < 8; ++r) {
        int row0 = m0 +  0 + r + 8 * half;
        int row1 = m0 + 16 + r + 8 * half;
        int col0 = n0 +  0 + lane_lo;
        int col1 = n0 + 16 + lane_lo;
        size_t i00 = (size_t)row0 * CLOTH_CH + col0;
        size_t i01 = (size_t)row0 * CLOTH_CH + col1;
        size_t i10 = (size_t)row1 * CLOTH_CH + col0;
        size_t i11 = (size_t)row1 * CLOTH_CH + col1;
        out[i00] = c00[r] * csum[i00];
        out[i01] = c01[r] * csum[i01];
        out[i10] = c10[r] * csum[i10];
        out[i11] = c11[r] * csum[i11];
    }
}

// ---------------------------------------------------------------------------
// Launch
// ---------------------------------------------------------------------------
extern "C" void kernel_launch(void* const* d_in, const int* in_sizes, int n_in,
                              void* d_out, int out_size, void* d_ws, size_t ws_size,
                              hipStream_t stream)
{
    const float* cloth = (const float*)d_in[0];   // (128, 64, 1024) f32
    const float* body  = (const float*)d_in[1];   // (128, 1024)     f32
    const float* f     = (const float*)d_in[2];   // (1024, 1024)    f32
    float*       out   = (float*)d_out;           // (128, 1024)     f32
    float*       csum  = (float*)d_ws;            // 128*1024*4 = 512 KB scratch

    // Kernel 1: 32768 threads -> 128 blocks of 256
    cloth_csum_kernel<<<(BATCH * CLOTH_CH / 4 + 255) / 256, 256, 0, stream>>>(cloth, csum);

    // Kernel 2: 128 waves (4 M-groups x 32 N-groups), 8 waves/block -> 16 blocks
    fused_gemm_wmma_kernel<<<16, 256, 0, stream>>>(body, f, csum, out);
}